// SpikingLateralLine_16673063043189
// MI455X (gfx1250) — compile-verified
//
#include <hip/hip_runtime.h>
#include <math.h>

#define BATCH   4096
#define N_NEU   1024
#define NSTEPS  10

typedef float v2f __attribute__((ext_vector_type(2)));
typedef float v8f __attribute__((ext_vector_type(8)));

// ---------------------------------------------------------------------------
// Kernel 1: per-batch geometry -> writes out[:, {0,1,2,5}] and per-batch drive
// currents into ws[b*3 + {0:ant_lo_half, 1:ant_hi_half, 2:post}]
// ---------------------------------------------------------------------------
__global__ void lateral_geom_kernel(const float* __restrict__ fx,
                                    const float* __restrict__ fy,
                                    const float* __restrict__ fh,
                                    const float* __restrict__ px,
                                    const float* __restrict__ py,
                                    const float* __restrict__ pvx,
                                    const float* __restrict__ pvy,
                                    float* __restrict__ out,
                                    float* __restrict__ ws) {
    int b = blockIdx.x * blockDim.x + threadIdx.x;
    if (b >= BATCH) return;

    float dx = px[b] - fx[b];
    float dy = py[b] - fy[b];
    float dist = sqrtf(dx * dx + dy * dy) + 1e-6f;
    bool in_range = dist <= 150.0f;

    float prox = in_range ? fmaxf(0.0f, 1.0f - dist * (1.0f / 150.0f)) : 0.0f;
    float angle_to = atan2f(dy, dx);
    float d = angle_to - fh[b];
    float rel = atan2f(sinf(d), cosf(d));

    float vx = pvx[b], vy = pvy[b];
    float speed = sqrtf(vx * vx + vy * vy);
    float flow_mag = in_range ? fminf(1.0f, speed * prox * 0.1f) : 0.0f;
    float flow_dir = in_range ? sinf(rel) : 0.0f;

    float crel = cosf(rel);
    float frontal = fmaxf(0.0f, crel);
    float caudal  = fmaxf(0.0f, -crel);

    float base_ant = prox * frontal * 15.0f;
    // boost 1.5x where (flow_dir > 0) == (n < N/2)
    float i_ant_lo, i_ant_hi;
    if (flow_dir > 0.0f) { i_ant_lo = base_ant * 1.5f; i_ant_hi = base_ant; }
    else                 { i_ant_lo = base_ant;        i_ant_hi = base_ant * 1.5f; }
    if (!in_range) { i_ant_lo = 0.0f; i_ant_hi = 0.0f; }
    float i_post = in_range ? prox * caudal * 12.0f : 0.0f;

    ws[b * 3 + 0] = i_ant_lo;
    ws[b * 3 + 1] = i_ant_hi;
    ws[b * 3 + 2] = i_post;

    out[b * 6 + 0] = prox;
    out[b * 6 + 1] = flow_dir;
    out[b * 6 + 2] = flow_mag;
    out[b * 6 + 5] = in_range ? dist : 999.0f;
}

// ---------------------------------------------------------------------------
// Kernel 2: Izhikevich loop + mean-rate reduction.
// grid = 2*BATCH blocks (blockIdx&1: 0=anterior, 1=posterior), 256 threads,
// each thread owns 4 consecutive neurons (float4 noise loads, all 10 steps'
// loads issued up front so the wave is pure-streaming bound).
// Wave-level rate reduction done with V_WMMA_F32_16X16X4_F32 (ones-matmul).
// ---------------------------------------------------------------------------
__global__ void __launch_bounds__(256)
lateral_izh_kernel(const float* __restrict__ noise_ant,
                   const float* __restrict__ noise_post,
                   const float* __restrict__ ws,
                   float* __restrict__ out) {
    const int b   = blockIdx.x >> 1;
    const int pop = blockIdx.x & 1;
    const int tid = threadIdx.x;
    const int n0  = tid * 4;

    const float* __restrict__ noise = pop ? noise_post : noise_ant;

    float drive;
    if (pop) drive = ws[b * 3 + 2];
    else     drive = (n0 < (N_NEU / 2)) ? ws[b * 3 + 0] : ws[b * 3 + 1];
    const float Ibase = drive + (-2.0f);   // I_TONIC

    // Issue all 10 float4 noise loads before any compute (fills LOADcnt queue).
    const float* p = noise + (size_t)b * N_NEU + n0;
    float4 nz[NSTEPS];
#pragma unroll
    for (int t = 0; t < NSTEPS; ++t)
        nz[t] = *(const float4*)(p + (size_t)t * ((size_t)BATCH * N_NEU));

    float v[4], u[4], r[4];
#pragma unroll
    for (int j = 0; j < 4; ++j) { v[j] = -65.0f; u[j] = 0.2f * -65.0f; r[j] = 0.0f; }

#pragma unroll
    for (int t = 0; t < NSTEPS; ++t) {
        const float nv[4] = { nz[t].x, nz[t].y, nz[t].z, nz[t].w };
#pragma unroll
        for (int j = 0; j < 4; ++j) {
            float I  = fmaf(nv[j], 0.5f, Ibase);
            float vj = v[j], uj = u[j];
            float vn = vj + (0.04f * vj * vj + 5.0f * vj + 140.0f - uj + I);
            float un = uj + 0.02f * (0.2f * vj - uj);
            bool spike = vn >= 30.0f;
            v[j] = spike ? -65.0f : vn;
            u[j] = spike ? un + 8.0f : un;
            r[j] = 0.95f * r[j] + (spike ? 0.05f : 0.0f);
        }
    }
    float partial = (r[0] + r[1]) + (r[2] + r[3]);

    // ---- exact f32 wave32 reduction via WMMA ones-matmul ----
    // A (16x4): VGPR0 = partial (per-lane), VGPR1 = 0  => row M holds
    //           {p(M), p(M+16)} in its two populated K slots.
    // B (4x16): all ones.
    // D[M][N] = p(M) + p(M+16) for every N.
    v2f a;  a[0] = partial; a[1] = 0.0f;
    v2f bm; bm[0] = 1.0f;   bm[1] = 1.0f;
    v8f c = {};
    c = __builtin_amdgcn_wmma_f32_16x16x4_f32(
            /*neg_a=*/false, a, /*neg_b=*/false, bm,
            /*c_mod=*/(short)0, c, /*reuse_a=*/false, /*reuse_b=*/false);
    // Each lane's 8 accumulator VGPRs cover 8 distinct rows; lane L and
    // lane L^16 cover complementary row sets -> sum + one xor-shuffle = total.
    float s = ((c[0] + c[1]) + (c[2] + c[3])) + ((c[4] + c[5]) + (c[6] + c[7]));
    s += __shfl_xor(s, 16, 32);

    __shared__ float wave_sum[8];
    const int wave = tid >> 5;
    const int lane = tid & 31;
    if (lane == 0) wave_sum[wave] = s;
    __syncthreads();
    if (tid == 0) {
        float tot = 0.0f;
#pragma unroll
        for (int w = 0; w < 8; ++w) tot += wave_sum[w];
        out[b * 6 + 3 + pop] = tot * (1.0f / N_NEU);
    }
}

// ---------------------------------------------------------------------------
extern "C" void kernel_launch(void* const* d_in, const int* in_sizes, int n_in,
                              void* d_out, int out_size, void* d_ws, size_t ws_size,
                              hipStream_t stream) {
    const float* fx  = (const float*)d_in[0];
    const float* fy  = (const float*)d_in[1];
    const float* fh  = (const float*)d_in[2];
    const float* px  = (const float*)d_in[3];
    const float* py  = (const float*)d_in[4];
    const float* pvx = (const float*)d_in[5];
    const float* pvy = (const float*)d_in[6];
    const float* noise_ant  = (const float*)d_in[7];
    const float* noise_post = (const float*)d_in[8];
    float* out = (float*)d_out;
    float* ws  = (float*)d_ws;

    lateral_geom_kernel<<<BATCH / 256, 256, 0, stream>>>(
        fx, fy, fh, px, py, pvx, pvy, out, ws);

    lateral_izh_kernel<<<2 * BATCH, 256, 0, stream>>>(
        noise_ant, noise_post, ws, out);
}